// Embedding_79061757984858
// MI455X (gfx1250) — compile-verified
//
#include <hip/hip_runtime.h>

#define SEQ   131072
#define DEPTH 512
#define VOCAB 118

#define W_FLOATS (VOCAB * DEPTH)          // 60416 floats = 236 KB
#define W_VEC4   (W_FLOATS / 4)           // 15104 float4s
#define BLOCK    512
#define NBLOCKS  512
#define ROWS_PER_ITER (BLOCK * 4 / DEPTH) // 4 rows per block iteration

typedef float v4f __attribute__((ext_vector_type(4)));

__global__ __launch_bounds__(BLOCK)
void embed_pe_kernel(const int* __restrict__ idx,
                     const float* __restrict__ W,
                     float* __restrict__ out)
{
    __shared__ float shW[W_FLOATS];  // whole embedding table, 236 KB (<320 KB/WGP)
    __shared__ float shF[256];       // inverse frequencies 10000^(-j/256)

    const int tid = threadIdx.x;

    // ---- Stage W into LDS with CDNA5 async global->LDS DMA (ASYNCcnt path) ----
    {
        const unsigned lds_base = (unsigned)(size_t)&shW[0]; // low 32 bits = LDS byte offset
        for (int i = tid; i < W_VEC4; i += BLOCK) {
            unsigned lds_addr = lds_base + (unsigned)i * 16u;
            unsigned goff     = (unsigned)i * 16u;
            asm volatile("global_load_async_to_lds_b128 %0, %1, %2"
                         :: "v"(lds_addr), "v"(goff), "s"(W)
                         : "memory");
        }
    }

    // invfreq[j] = 10000^(-j/256) = exp2(-j * log2(10000)/256)
    if (tid < 256)
        shF[tid] = __builtin_amdgcn_exp2f(-0.05190512648261504f * (float)tid);

    asm volatile("s_wait_asynccnt 0x0" ::: "memory");
    __syncthreads();

    const int c  = tid & (DEPTH / 4 - 1);  // float4 column index 0..127
    const int rs = tid >> 7;               // row slot within block 0..3
    const int j0 = 2 * c;                  // sin/cos pair for cols 4c,4c+1
    const int j1 = 2 * c + 1;              // pair for cols 4c+2,4c+3 (j==255 -> zeros)
    const float if0 = shF[j0];
    const float if1 = shF[j1];
    const double INV2PI = 0.15915494309189535;
    const bool tailzero = (j1 == DEPTH / 2 - 1); // columns 510,511 stay zero

    for (int rbase = (int)blockIdx.x * ROWS_PER_ITER; rbase < SEQ;
         rbase += (int)gridDim.x * ROWS_PER_ITER) {
        const int row = rbase + rs;
        const int e   = idx[row];

        // gather embedding row slice from LDS (b128)
        v4f w = *(const v4f*)&shW[e * DEPTH + 4 * c];

        // angle in fp32 (as reference), range-reduce in fp64 turns, hardware sin/cos
        const float rf = (float)row;
        float ang0 = rf * if0;
        float ang1 = rf * if1;

        double t0 = (double)ang0 * INV2PI; t0 -= floor(t0);
        double t1 = (double)ang1 * INV2PI; t1 -= floor(t1);

        float s0 = __builtin_amdgcn_sinf((float)t0);
        float c0 = __builtin_amdgcn_cosf((float)t0);
        float s1 = __builtin_amdgcn_sinf((float)t1);
        float c1 = __builtin_amdgcn_cosf((float)t1);
        if (tailzero) { s1 = 0.0f; c1 = 0.0f; }

        v4f o;
        o.x = w.x + s0;
        o.y = w.y + c0;
        o.z = w.z + s1;
        o.w = w.w + c1;

        // streamed output: non-temporal b128 store, keep L2 for W/idx
        __builtin_nontemporal_store(o, (v4f*)&out[row * DEPTH + 4 * c]);
    }
}

extern "C" void kernel_launch(void* const* d_in, const int* in_sizes, int n_in,
                              void* d_out, int out_size, void* d_ws, size_t ws_size,
                              hipStream_t stream) {
    (void)in_sizes; (void)n_in; (void)d_ws; (void)ws_size; (void)out_size;
    const int*   idx = (const int*)d_in[0];
    const float* W   = (const float*)d_in[1];
    float*       out = (float*)d_out;
    embed_pe_kernel<<<NBLOCKS, BLOCK, 0, stream>>>(idx, W, out);
}